// AnchorPatchPooling_27324581937299
// MI455X (gfx1250) — compile-verified
//
#include <hip/hip_runtime.h>
#include <stdint.h>

// Problem constants (match reference)
#define N_ 32
#define C_ 256
#define K_ 8192
#define P_ 16
#define CT 16          // channel tile per block (== WMMA M)
#define NW 8           // waves per block (wave32)
#define THREADS 256

typedef float v2f __attribute__((ext_vector_type(2)));
typedef float v8f __attribute__((ext_vector_type(8)));

// One block per (n, 16-channel tile). The scatter-sum runs on the matrix unit
// as out[16c x 16p] += A(16c x 4k) * B(4k x 16p) with B the masked one-hot of
// part labels; the scatter-max runs on ds_max_num_f32 into an LDS tile using
// the same A-fragment registers.
__global__ __launch_bounds__(THREADS)
void AnchorPatchPooling_kernel(const float* __restrict__ feats,
                               const int*  __restrict__ labels,
                               const int*  __restrict__ vmask,
                               float* __restrict__ out)
{
    __shared__ uint32_t lv[K_];            // packed {label[31:16], onehot[15:0]}
    __shared__ float sumTile[CT * P_];     // 16c x 16p partial sums
    __shared__ float maxTile[CT * 17];     // padded stride 17 -> conflict-free ds_max
    __shared__ float cntBin[P_];           // pooled_count[n][p] (block-local copy)
    __shared__ float patchBin[P_];         // patch_count[p]     (block-local copy)

    const int t   = threadIdx.x;
    const int blk = blockIdx.x;
    const int n   = blk >> 4;              // 16 channel tiles per sample
    const int c0  = (blk & 15) * CT;

    // ---- init LDS ----
    if (t < CT * P_) sumTile[t] = 0.0f;
    for (int i = t; i < CT * 17; i += THREADS) maxTile[i] = -100.0f;  // MAX_INIT
    if (t < P_) { cntBin[t] = 0.0f; patchBin[t] = 0.0f; }
    __syncthreads();

    // ---- stage labels+mask, build per-part counts (ds_add_f32) ----
    for (int k = t; k < K_; k += THREADS) {
        uint32_t lab = (uint32_t)labels[k] & (P_ - 1);
        int v = vmask[(size_t)n * K_ + k];
        // low 16 bits: masked one-hot of the label; high bits: the label itself
        lv[k] = (lab << 16) | (v ? (1u << lab) : 0u);
        atomicAdd(&patchBin[lab], 1.0f);
        if (v) atomicAdd(&cntBin[lab], 1.0f);
    }
    __syncthreads();

    // ---- main loop: WMMA scatter-sum + ds_max scatter-max ----
    const int lane = t & 31;
    const int wv   = t >> 5;
    const int col  = lane & 15;            // WMMA row (A) / part column (B,C)
    const int hi   = lane >> 4;            // k half selector within the chunk

    const int kStart     = wv * (K_ / NW); // each wave owns a contiguous K slice
    const float*    aPtr  = feats + ((size_t)(n * C_ + c0 + col)) * K_ + kStart + 2 * hi;
    const uint32_t* lvPtr = &lv[kStart + 2 * hi];
    // byte address of this lane's padded max-tile row (generic addr low 32 = LDS offset)
    const uint32_t  maxRow = (uint32_t)(uintptr_t)(&maxTile[0]) + (uint32_t)(col * 17) * 4u;

    v8f acc = {};  // 16x16 f32 accumulator (8 VGPRs, wave-striped)

    #pragma unroll 8
    for (int i = 0; i < K_ / NW; i += 4) {
        uint32_t w0 = lvPtr[i];
        uint32_t w1 = lvPtr[i + 1];

        // B fragment: one bit-extract per element from the masked one-hot
        v2f b;
        b.x = (float)((w0 >> col) & 1u);
        b.y = (float)((w1 >> col) & 1u);

        // A fragment: 16 channels x 4 k, 8 bytes per lane, stream-coalesced
        v2f a = *(const v2f*)(aPtr + i);

        acc = __builtin_amdgcn_wmma_f32_16x16x4_f32(
            /*neg_a=*/false, a, /*neg_b=*/false, b,
            /*c_mod=*/(short)0, acc, /*reuse_a=*/false, /*reuse_b=*/false);

        // scatter-max of the raw (unmasked) values into the LDS tile.
        // No "memory" clobber: these only touch maxTile, which is read only
        // after the explicit s_wait_dscnt + barrier below; this lets the
        // compiler keep global loads in flight across the asm.
        uint32_t addr0 = maxRow + ((w0 >> 16) << 2);
        uint32_t addr1 = maxRow + ((w1 >> 16) << 2);
        asm volatile("ds_max_num_f32 %0, %1" :: "v"(addr0), "v"(a.x));
        asm volatile("ds_max_num_f32 %0, %1" :: "v"(addr1), "v"(a.y));
    }

    // ---- reduce wave-private C tiles into the shared sum tile ----
    #pragma unroll
    for (int j = 0; j < 8; ++j) {
        int m = j + 8 * hi;                          // C layout: M = vgpr + 8*(lane/16)
        atomicAdd(&sumTile[m * P_ + col], acc[j]);
    }
    // Inline-asm DS ops are invisible to the compiler's counter tracking:
    // drain them explicitly before the barrier.
    asm volatile("s_wait_dscnt 0x0" ::: "memory");
    __syncthreads();

    // ---- epilogue: mean + gated max, one output element per thread ----
    {
        int m = t >> 4, p = t & 15;
        float s    = sumTile[m * P_ + p];
        float cntp = cntBin[p];
        float mean = s / fmaxf(cntp, 1.0f);
        float mx   = (patchBin[p] > 0.0f) ? maxTile[m * 17 + p] : 0.0f;
        out[((size_t)(n * C_ + c0 + m)) * P_ + p] = mean + mx;
    }
}

extern "C" void kernel_launch(void* const* d_in, const int* in_sizes, int n_in,
                              void* d_out, int out_size, void* d_ws, size_t ws_size,
                              hipStream_t stream) {
    (void)in_sizes; (void)n_in; (void)d_ws; (void)ws_size; (void)out_size;
    const float* feats  = (const float*)d_in[0];
    const int*   labels = (const int*)d_in[1];   // part_labels (int)
    const int*   vmask  = (const int*)d_in[2];   // valid_mask  (int 0/1)
    float*       out    = (float*)d_out;         // [N, C, PARTS] f32

    dim3 grid(N_ * (C_ / CT));   // 512 blocks
    dim3 block(THREADS);         // 8 wave32s
    AnchorPatchPooling_kernel<<<grid, block, 0, stream>>>(feats, labels, vmask, out);
}